// Solver_43284680409442
// MI455X (gfx1250) — compile-verified
//
#include <hip/hip_runtime.h>

// ---------------- problem constants ----------------
#define B_    32
#define S_    1024
#define E_    1024
#define HID_  2048
#define KSLOT 200
#define NCE_T 0.1f

typedef float v2f __attribute__((ext_vector_type(2)));
typedef float v8f __attribute__((ext_vector_type(8)));
typedef int   v4i __attribute__((ext_vector_type(4)));

// ---- CDNA5 async global->LDS path (guarded; sync float4 fallback) ----
#if __has_builtin(__builtin_amdgcn_global_load_async_to_lds_b128) && \
    __has_builtin(__builtin_amdgcn_s_wait_asynccnt)
#define USE_ASYNC 1
typedef __attribute__((address_space(1))) v4i* as1_v4i_ptr;   // global (printed '__device__')
typedef __attribute__((address_space(3))) v4i* as3_v4i_ptr;   // LDS    (printed '__shared__')
#define GPTR(p) ((as1_v4i_ptr)(void*)(p))
#define LPTR(p) ((as3_v4i_ptr)(void*)(p))
#else
#define USE_ASYNC 0
#endif

__device__ __forceinline__ float gumbelf(float u) {
  u = fminf(fmaxf(u, 1e-10f), 1.0f - 1e-10f);
  return -logf(-logf(u));
}

// =====================================================================
// WMMA fp32 GEMM:  C[M,N] = A[M,K] @ B[K,N] (+bias) (opt ReLU)
// Requirements: M % 128 == 0, K % 16 == 0, ldb >= round-up of N to tile,
//               lda/ldb multiples of 4.  N bounds-checked on store.
// Block: 256 thr (8 waves). Block tile 128x128; wave tile 32x64 (2x4 WMMA).
// Double-buffered LDS stages fed by GLOBAL_LOAD_ASYNC_TO_LDS_B128
// (4 async instr per wave per stage; ASYNCcnt completes in order).
// =====================================================================
__device__ __forceinline__ void stage_issue(
    const float* __restrict__ A, const float* __restrict__ B,
    float* __restrict__ As, float* __restrict__ Bs,
    int tid, int bm0, int bn0, int k0, int lda, int ldb)
{
  // A tile (128 x 16): 512 float4, 2 per thread
  for (int i = 0; i < 2; i++) {
    int idx = tid * 2 + i;
    int r = idx >> 2, c4 = idx & 3;
    const float* gp = &A[(size_t)(bm0 + r) * lda + k0 + c4 * 4];
    float* lp = &As[r * 20 + c4 * 4];            // byte addr 80r+16c4: 16B aligned
#if USE_ASYNC
    __builtin_amdgcn_global_load_async_to_lds_b128(GPTR(gp), LPTR(lp), 0, 0);
#else
    const float4 v = *reinterpret_cast<const float4*>(gp);
    lp[0] = v.x; lp[1] = v.y; lp[2] = v.z; lp[3] = v.w;
#endif
  }
  // B tile (16 x 128): 512 float4, 2 per thread
  for (int i = 0; i < 2; i++) {
    int idx = tid * 2 + i;
    int r = idx >> 5, c4 = idx & 31;
    const float* gp = &B[(size_t)(k0 + r) * ldb + bn0 + c4 * 4];
    float* lp = &Bs[r * 132 + c4 * 4];           // byte addr 528r+16c4: 16B aligned
#if USE_ASYNC
    __builtin_amdgcn_global_load_async_to_lds_b128(GPTR(gp), LPTR(lp), 0, 0);
#else
    const float4 v = *reinterpret_cast<const float4*>(gp);
    lp[0] = v.x; lp[1] = v.y; lp[2] = v.z; lp[3] = v.w;
#endif
  }
}

__global__ __launch_bounds__(256) void gemm_wmma_f32(
    const float* __restrict__ A, const float* __restrict__ B,
    const float* __restrict__ bias, float* __restrict__ C,
    int M, int N, int K, int lda, int ldb, int ldc, int relu)
{
  __shared__ float As[2][128 * 20];   // 128 rows x 16 k (stride 20), double buffered
  __shared__ float Bs[2][16 * 132];   // 16 k rows x 128 n (stride 132)

  const int tid  = threadIdx.x;
  const int lane = tid & 31;
  const int wv   = tid >> 5;
  const int wm0  = (wv >> 1) * 32;   // 4 waves in M
  const int wn0  = (wv & 1) * 64;    // 2 waves in N
  const int bm0  = blockIdx.y * 128;
  const int bn0  = blockIdx.x * 128;
  const int lo   = (lane < 16) ? 1 : 0;
  const int l15  = lane & 15;

  v8f acc[2][4];
  for (int tm = 0; tm < 2; tm++)
    for (int tn = 0; tn < 4; tn++)
      for (int j = 0; j < 8; j++) acc[tm][tn][j] = 0.0f;

  // prologue: stage 0 in flight
  stage_issue(A, B, As[0], Bs[0], tid, bm0, bn0, 0, lda, ldb);

  int it = 0;
  for (int k0 = 0; k0 < K; k0 += 16, it ^= 1) {
    // issue next stage while current computes
    if (k0 + 16 < K)
      stage_issue(A, B, As[it ^ 1], Bs[it ^ 1], tid, bm0, bn0, k0 + 16, lda, ldb);
#if USE_ASYNC
    // async loads complete in order: <=4 outstanding  ==> current stage landed
    if (k0 + 16 < K) __builtin_amdgcn_s_wait_asynccnt(4);
    else             __builtin_amdgcn_s_wait_asynccnt(0);
#endif
    __syncthreads();                 // all waves' current-stage data visible

    const float* __restrict__ Asb = As[it];
    const float* __restrict__ Bsb = Bs[it];
    for (int kk = 0; kk < 16; kk += 4) {
      v2f a[2], b[4];
      for (int tm = 0; tm < 2; tm++) {
        int row = wm0 + tm * 16 + l15;
        a[tm].x = Asb[row * 20 + kk + (lo ? 0 : 2)];
        a[tm].y = Asb[row * 20 + kk + (lo ? 1 : 3)];
      }
      for (int tn = 0; tn < 4; tn++) {
        int col = wn0 + tn * 16 + l15;
        b[tn].x = Bsb[(kk + (lo ? 0 : 2)) * 132 + col];
        b[tn].y = Bsb[(kk + (lo ? 1 : 3)) * 132 + col];
      }
      for (int tm = 0; tm < 2; tm++)
        for (int tn = 0; tn < 4; tn++)
          acc[tm][tn] = __builtin_amdgcn_wmma_f32_16x16x4_f32(
              false, a[tm], false, b[tn], (short)0, acc[tm][tn], false, false);
    }
    __syncthreads();                 // done reading buf[it] before it is refilled
  }

  // ---- epilogue: bias / relu / bounded store ----
  const int hi8 = (lane >= 16) ? 8 : 0;
  for (int tm = 0; tm < 2; tm++)
    for (int tn = 0; tn < 4; tn++) {
      int col = bn0 + wn0 + tn * 16 + l15;
      if (col < N) {
        float bv = bias ? bias[col] : 0.0f;
        for (int j = 0; j < 8; j++) {
          int row = bm0 + wm0 + tm * 16 + hi8 + j;
          float v = acc[tm][tn][j] + bv;
          if (relu) v = fmaxf(v, 0.0f);
          C[(size_t)row * ldc + col] = v;
        }
      }
    }
}

// ---------------- container^T with zero pad to 256 cols ----------------
__global__ void transpose_container(const float* __restrict__ c, float* __restrict__ ct) {
  int gid = blockIdx.x * 256 + threadIdx.x;   // E_*256 total
  if (gid >= E_ * 256) return;
  int e = gid >> 8, k = gid & 255;
  ct[gid] = (k < KSLOT) ? c[k * E_ + e] : 0.0f;
}

// ---------------- argmax over S of (masked sim + gumbel) ----------------
__global__ __launch_bounds__(256) void argmax_y1(
    const float* __restrict__ sim /*[B*S,200]*/, const float* __restrict__ u1,
    const int* __restrict__ trans_len, int* __restrict__ sel_idx)
{
  __shared__ float bv[256]; __shared__ int bi[256];
  int bk = blockIdx.x;             // b*KSLOT + k
  int b = bk / KSLOT, k = bk % KSLOT;
  int len = trans_len[b];
  float bestv = -3.4e38f; int besti = 0x7fffffff;
  for (int s = threadIdx.x; s < S_; s += 256) {
    float g  = gumbelf(u1[(size_t)bk * S_ + s]);
    float sv = (s < len) ? sim[(size_t)(b * S_ + s) * KSLOT + k] : -3.4e38f;
    float y  = sv + g;             // TAU = 1
    if (y > bestv || (y == bestv && s < besti)) { bestv = y; besti = s; }
  }
  bv[threadIdx.x] = bestv; bi[threadIdx.x] = besti; __syncthreads();
  for (int o = 128; o > 0; o >>= 1) {
    if (threadIdx.x < o) {
      if (bv[threadIdx.x + o] > bv[threadIdx.x] ||
          (bv[threadIdx.x + o] == bv[threadIdx.x] && bi[threadIdx.x + o] < bi[threadIdx.x])) {
        bv[threadIdx.x] = bv[threadIdx.x + o]; bi[threadIdx.x] = bi[threadIdx.x + o];
      }
    }
    __syncthreads();
  }
  if (threadIdx.x == 0) sel_idx[bk] = bi[0];
}

// ---------------- gather: joint = [container | trans[sel]] ----------------
__global__ __launch_bounds__(256) void build_joint(
    const float* __restrict__ trans, const float* __restrict__ container,
    const int* __restrict__ sel_idx, float* __restrict__ joint, float* __restrict__ selsent)
{
  int bk = blockIdx.x; int b = bk / KSLOT, k = bk % KSLOT;
  int s = sel_idx[bk];
  const float* tr = trans + (size_t)(b * S_ + s) * E_;
  const float* cc = container + (size_t)k * E_;
  float* jr = joint + (size_t)bk * 2 * E_;
  float* ss = selsent + (size_t)bk * E_;
  for (int e = threadIdx.x; e < E_; e += 256) {
    float tv = tr[e];
    jr[e] = cc[e]; jr[E_ + e] = tv; ss[e] = tv;
  }
}

// ---------------- in-place row LayerNorm ----------------
__global__ __launch_bounds__(256) void layernorm_rows(
    float* __restrict__ X, const float* __restrict__ g, const float* __restrict__ b, int D)
{
  __shared__ float red[256];
  __shared__ float s_mean, s_inv;
  int row = blockIdx.x, tid = threadIdx.x;
  float* x = X + (size_t)row * D;
  float s = 0.f, s2 = 0.f;
  for (int i = tid; i < D; i += 256) { float v = x[i]; s += v; s2 += v * v; }
  red[tid] = s; __syncthreads();
  for (int o = 128; o > 0; o >>= 1) { if (tid < o) red[tid] += red[tid + o]; __syncthreads(); }
  if (tid == 0) s_mean = red[0] / D;
  __syncthreads();
  red[tid] = s2; __syncthreads();
  for (int o = 128; o > 0; o >>= 1) { if (tid < o) red[tid] += red[tid + o]; __syncthreads(); }
  if (tid == 0) { float var = red[0] / D - s_mean * s_mean; s_inv = 1.0f / sqrtf(var + 1e-5f); }
  __syncthreads();
  float mu = s_mean, inv = s_inv;
  for (int i = tid; i < D; i += 256) x[i] = (x[i] - mu) * inv * g[i] + b[i];
}

// ---------------- out[r] = dot(H[r,:], w) + b0 ----------------
__global__ __launch_bounds__(256) void rowdot_kernel(
    const float* __restrict__ H, const float* __restrict__ w,
    const float* __restrict__ b0, float* __restrict__ outp, int D)
{
  __shared__ float red[256];
  int r = blockIdx.x, tid = threadIdx.x;
  const float* h = H + (size_t)r * D;
  float s = 0.f;
  for (int i = tid; i < D; i += 256) s += h[i] * w[i];
  red[tid] = s; __syncthreads();
  for (int o = 128; o > 0; o >>= 1) { if (tid < o) red[tid] += red[tid + o]; __syncthreads(); }
  if (tid == 0) outp[r] = red[0] + b0[0];
}

// ---------------- concise KL ----------------
__global__ __launch_bounds__(256) void concise_kernel(const float* __restrict__ out, float* d_out) {
  __shared__ float red[256];
  __shared__ float s_b;
  int tid = threadIdx.x;
  float total = 0.f;
  const float logK = logf(1.0f / (float)KSLOT);
  for (int b = 0; b < B_; b++) {
    const float* o = out + b * KSLOT;
    float m = -3.4e38f;
    for (int k = tid; k < KSLOT; k += 256) m = fmaxf(m, o[k]);
    red[tid] = m; __syncthreads();
    for (int q = 128; q > 0; q >>= 1) { if (tid < q) red[tid] = fmaxf(red[tid], red[tid + q]); __syncthreads(); }
    if (tid == 0) s_b = red[0];
    __syncthreads();
    float mx = s_b, s = 0.f;
    for (int k = tid; k < KSLOT; k += 256) s += expf(o[k] - mx);
    red[tid] = s; __syncthreads();
    for (int q = 128; q > 0; q >>= 1) { if (tid < q) red[tid] += red[tid + q]; __syncthreads(); }
    if (tid == 0) s_b = mx + logf(red[0]);
    __syncthreads();
    float lse = s_b, c = 0.f;
    for (int k = tid; k < KSLOT; k += 256) c += (logK - (o[k] - lse));
    red[tid] = c; __syncthreads();
    for (int q = 128; q > 0; q >>= 1) { if (tid < q) red[tid] += red[tid + q]; __syncthreads(); }
    if (tid == 0) total += red[0] / (float)KSLOT;
    __syncthreads();
  }
  if (tid == 0) d_out[1] = total / (float)B_;
}

// ---------------- per-batch variable-K top-k (khot) ----------------
__global__ __launch_bounds__(256) void khot_kernel(
    const float* __restrict__ out, const float* __restrict__ u2,
    const int* __restrict__ trans_len, float* __restrict__ khot, int* __restrict__ sel_len)
{
  __shared__ float y[256];
  int b = blockIdx.x, tid = threadIdx.x;
  for (int k = tid; k < KSLOT; k += 256) y[k] = out[b * KSLOT + k] + gumbelf(u2[b * KSLOT + k]);
  __syncthreads();
  int len = (int)ceilf((float)trans_len[b] * 0.15f) + 5;
  if (len > KSLOT) len = KSLOT;
  if (tid == 0) sel_len[b] = len;
  for (int k = tid; k < KSLOT; k += 256) {
    float yk = y[k]; int r = 0;
    for (int j = 0; j < KSLOT; j++)
      if (y[j] > yk || (y[j] == yk && j < k)) r++;
    khot[b * KSLOT + k] = (r < len) ? 1.0f : 0.0f;
  }
}

// ---------------- misc small kernels ----------------
__global__ void zero_f32(float* __restrict__ p, int n) {
  int gid = blockIdx.x * 256 + threadIdx.x;
  if (gid < n) p[gid] = 0.0f;
}

__global__ void resmask_scatter(const float* __restrict__ khot, const int* __restrict__ sel_idx,
                                float* __restrict__ resmask) {
  int bk = blockIdx.x * 256 + threadIdx.x;
  if (bk >= B_ * KSLOT) return;
  int b = bk / KSLOT;
  atomicAdd(&resmask[b * S_ + sel_idx[bk]], khot[bk]);
}

__global__ void mean_sel_kernel(const float* __restrict__ selsent, const float* __restrict__ khot,
                                const int* __restrict__ sel_len, float* __restrict__ outp) {
  int gid = blockIdx.x * 256 + threadIdx.x;
  if (gid >= B_ * E_) return;
  int b = gid / E_, e = gid % E_;
  float acc = 0.f;
  for (int k = 0; k < KSLOT; k++)
    acc += khot[b * KSLOT + k] * selsent[(size_t)(b * KSLOT + k) * E_ + e];
  outp[gid] = acc / (float)sel_len[b];
}

__global__ void ultimate_kernel(const float* __restrict__ pt, const float* __restrict__ khot,
                                const int* __restrict__ sel_idx, float* __restrict__ outp) {
  size_t gid = (size_t)blockIdx.x * 256 + threadIdx.x;
  if (gid >= (size_t)B_ * KSLOT * E_) return;
  int e = (int)(gid % E_); int bk = (int)(gid / E_); int b = bk / KSLOT;
  outp[gid] = khot[bk] * pt[((size_t)b * S_ + sel_idx[bk]) * E_ + e];
}

__global__ void mss_kernel(const float* __restrict__ selsent, float* __restrict__ mss) {
  int gid = blockIdx.x * 256 + threadIdx.x;
  if (gid >= KSLOT * E_) return;
  int k = gid / E_, e = gid % E_;
  float acc = 0.f;
  for (int b = 0; b < B_; b++) acc += selsent[(size_t)(b * KSLOT + k) * E_ + e];
  mss[gid] = acc / (float)B_;
}

// small GEMM, one block per row (M small): C[row,:] = A[row,:]@W + bias
__global__ __launch_bounds__(256) void small_gemm(
    const float* __restrict__ A, const float* __restrict__ W,
    const float* __restrict__ bias, float* __restrict__ C, int K, int N)
{
  __shared__ float a[1024];
  int row = blockIdx.x, tid = threadIdx.x;
  for (int k = tid; k < K; k += 256) a[k] = A[(size_t)row * K + k];
  __syncthreads();
  for (int n = tid; n < N; n += 256) {
    float acc = bias[n];
    for (int k = 0; k < K; k++) acc += a[k] * W[(size_t)k * N + n];
    C[(size_t)row * N + n] = acc;
  }
}

__global__ __launch_bounds__(256) void l2norm_rows(
    const float* __restrict__ src, float* __restrict__ dst, int D)
{
  __shared__ float red[256];
  int row = blockIdx.x, tid = threadIdx.x;
  const float* x = src + (size_t)row * D;
  float s = 0.f;
  for (int i = tid; i < D; i += 256) { float v = x[i]; s += v * v; }
  red[tid] = s; __syncthreads();
  for (int o = 128; o > 0; o >>= 1) { if (tid < o) red[tid] += red[tid + o]; __syncthreads(); }
  float inv = 1.0f / (sqrtf(red[0]) + 1e-12f);
  for (int i = tid; i < D; i += 256) dst[(size_t)row * D + i] = x[i] * inv;
}

__global__ __launch_bounds__(256) void nce_kernel(
    const float* __restrict__ a, const float* __restrict__ p, float* d_out)
{
  __shared__ float an[B_], pn[B_], lg[B_ * B_], rl[B_];
  int tid = threadIdx.x;
  if (tid < B_) {
    float s = 0.f; const float* x = a + (size_t)tid * E_;
    for (int e = 0; e < E_; e++) s += x[e] * x[e];
    an[tid] = sqrtf(s) + 1e-12f;
  } else if (tid < 2 * B_) {
    int r = tid - B_; float s = 0.f; const float* x = p + (size_t)r * E_;
    for (int e = 0; e < E_; e++) s += x[e] * x[e];
    pn[r] = sqrtf(s) + 1e-12f;
  }
  __syncthreads();
  for (int t = tid; t < B_ * B_; t += 256) {
    int i = t / B_, j = t % B_;
    float acc = 0.f;
    const float* xi = a + (size_t)i * E_; const float* xj = p + (size_t)j * E_;
    for (int e = 0; e < E_; e++) acc += xi[e] * xj[e];
    lg[t] = acc / (an[i] * pn[j]) / NCE_T;
  }
  __syncthreads();
  if (tid < B_) {
    float m = -3.4e38f;
    for (int j = 0; j < B_; j++) m = fmaxf(m, lg[tid * B_ + j]);
    float s = 0.f;
    for (int j = 0; j < B_; j++) s += expf(lg[tid * B_ + j] - m);
    rl[tid] = m + logf(s) - lg[tid * B_ + tid];
  }
  __syncthreads();
  if (tid == 0) { float t = 0.f; for (int i = 0; i < B_; i++) t += rl[i]; d_out[0] = t / B_; }
}

__global__ __launch_bounds__(256) void gram_kernel(const float* __restrict__ chat, float* __restrict__ G) {
  __shared__ float ci[E_];
  int i = blockIdx.x, tid = threadIdx.x;
  for (int e = tid; e < E_; e += 256) ci[e] = chat[(size_t)i * E_ + e];
  __syncthreads();
  for (int j = tid; j < KSLOT; j += 256) {
    float acc = 0.f;
    const float* cj = chat + (size_t)j * E_;
    for (int e = 0; e < E_; e++) acc += ci[e] * cj[e];
    G[i * KSLOT + j] = acc;
  }
}

__global__ __launch_bounds__(256) void redundancy_kernel(
    const float* __restrict__ chat, const float* __restrict__ phat,
    const float* __restrict__ G, float* d_out)
{
  __shared__ float red[256];
  int tid = threadIdx.x;
  float loss = 0.f;
  if (tid < KSLOT) {
    int n = tid;
    float pos = 0.f;
    for (int e = 0; e < E_; e++) pos += chat[(size_t)n * E_ + e] * phat[(size_t)n * E_ + e];
    float lp = pos / NCE_T;
    float m = lp;
    for (int j = 0; j < KSLOT; j++) if (j != n) m = fmaxf(m, G[n * KSLOT + j] / NCE_T);
    float s = expf(lp - m);
    for (int j = 0; j < KSLOT; j++) if (j != n) s += expf(G[n * KSLOT + j] / NCE_T - m);
    loss = (m + logf(s)) - lp;
  }
  red[tid] = loss; __syncthreads();
  for (int o = 128; o > 0; o >>= 1) { if (tid < o) red[tid] += red[tid + o]; __syncthreads(); }
  if (tid == 0) d_out[3] = red[0] / (float)KSLOT;
}

__global__ __launch_bounds__(256) void uniform_kernel(const float* __restrict__ G, float* d_out) {
  __shared__ float red[256];
  int tid = threadIdx.x;
  float s = 0.f; int p = 0;
  for (int i = 0; i < KSLOT; i++)
    for (int j = i + 1; j < KSLOT; j++) {
      if ((p & 255) == tid) {
        float d2 = G[i * KSLOT + i] + G[j * KSLOT + j] - 2.0f * G[i * KSLOT + j];
        s += expf(-2.0f * d2);
      }
      p++;
    }
  red[tid] = s; __syncthreads();
  for (int o = 128; o > 0; o >>= 1) { if (tid < o) red[tid] += red[tid + o]; __syncthreads(); }
  if (tid == 0) d_out[2] = logf(red[0] / (float)(KSLOT * (KSLOT - 1) / 2));
}

// =====================================================================
extern "C" void kernel_launch(void* const* d_in, const int* in_sizes, int n_in,
                              void* d_out, int out_size, void* d_ws, size_t ws_size,
                              hipStream_t stream) {
  (void)in_sizes; (void)n_in; (void)out_size; (void)ws_size;
  // ----- inputs -----
  const float* pt     = (const float*)d_in[0];   // padded_trans [B,S,E]
  const float* meanqa = (const float*)d_in[1];   // [B,E]
  const float* u1     = (const float*)d_in[2];   // [B,K,S]
  const float* u2     = (const float*)d_in[3];   // [B,K]
  const int*   tlen   = (const int*)  d_in[4];   // [B]
  // d_in[5], d_in[6] = mask/att_mask (recomputed from tlen)
  const float* container = (const float*)d_in[7];
  const float* sentW = (const float*)d_in[8];  const float* sentb = (const float*)d_in[9];
  const float* transW= (const float*)d_in[10]; const float* transb= (const float*)d_in[11];
  const float* qaW   = (const float*)d_in[12]; const float* qab   = (const float*)d_in[13];
  const float* lnIg  = (const float*)d_in[14]; const float* lnIb  = (const float*)d_in[15];
  const float* W1    = (const float*)d_in[16]; const float* b1    = (const float*)d_in[17];
  const float* ln1g  = (const float*)d_in[18]; const float* ln1b  = (const float*)d_in[19];
  const float* W2    = (const float*)d_in[20]; const float* b2    = (const float*)d_in[21];
  const float* ln2g  = (const float*)d_in[22]; const float* ln2b  = (const float*)d_in[23];
  const float* Wout  = (const float*)d_in[24]; const float* bout  = (const float*)d_in[25];

  // ----- workspace layout (floats); trans region reused for MLP acts,
  //       sim region reused for joint -----
  float* ws = (float*)d_ws;
  const size_t NTRANS = (size_t)B_ * S_ * E_;          // 33,554,432
  const size_t NJOINT = (size_t)B_ * KSLOT * 2 * E_;   // 13,107,200
  float* trans  = ws;                                   // dies after gather
  float* h_a    = ws;                                   // reuse of trans
  float* h_b    = ws + NJOINT;                          // still inside trans region
  float* simJ   = ws + NTRANS;                          // sim [B*S,200] then joint [6400,2048]
  float* ctT    = simJ + NJOINT;                        // [E,256] zero-padded
  float* selsent= ctT + (size_t)E_ * 256;               // [B*K,E]
  float* mss    = selsent + (size_t)B_ * KSLOT * E_;    // [K,E]
  float* chat   = mss  + (size_t)KSLOT * E_;
  float* phat   = chat + (size_t)KSLOT * E_;
  float* G      = phat + (size_t)KSLOT * E_;            // [K,K]
  float* outbk  = G + (size_t)KSLOT * KSLOT;            // [B,K]
  float* khot   = outbk + (size_t)B_ * KSLOT;           // [B,K]
  float* anc    = khot  + (size_t)B_ * KSLOT;           // [B,E]
  float* posv   = anc   + (size_t)B_ * E_;              // [B,E]
  int*   sel_idx= (int*)(posv + (size_t)B_ * E_);       // [B,K]
  int*   sel_len= sel_idx + (size_t)B_ * KSLOT;         // [B]

  // ----- output layout -----
  float* o = (float*)d_out;                      // [0]=nce [1]=concise [2]=uniform [3]=redund
  float* o_resmask = o + 4;                      // [B,S]
  float* o_meansel = o + 4 + (size_t)B_ * S_;    // [B,E]
  float* o_ult     = o + 4 + (size_t)B_ * S_ + (size_t)B_ * E_;  // [B,K,E]

  const int BK = B_ * KSLOT;  // 6400

  // 1) container^T (zero-padded to 256 columns)
  transpose_container<<<(E_ * 256) / 256, 256, 0, stream>>>(container, ctT);

  // 2) trans = padded_trans @ sent_W + sent_b        [32768,1024]
  gemm_wmma_f32<<<dim3(E_ / 128, (B_ * S_) / 128), 256, 0, stream>>>(
      pt, sentW, sentb, trans, B_ * S_, E_, E_, E_, E_, E_, 0);

  // 3) sim^T = trans @ container^T                   [32768,200]
  gemm_wmma_f32<<<dim3(2, (B_ * S_) / 128), 256, 0, stream>>>(
      trans, ctT, nullptr, simJ, B_ * S_, KSLOT, E_, E_, 256, KSLOT, 0);

  // 4) hard gumbel-softmax selection (argmax over S)
  argmax_y1<<<BK, 256, 0, stream>>>(simJ, u1, tlen, sel_idx);

  // 5) gather -> selected_sent, build joint (overwrites sim region)
  build_joint<<<BK, 256, 0, stream>>>(trans, container, sel_idx, simJ, selsent);

  // 6) ImportanceScore MLP
  layernorm_rows<<<BK, 256, 0, stream>>>(simJ, lnIg, lnIb, 2 * E_);
  gemm_wmma_f32<<<dim3(HID_ / 128, BK / 128), 256, 0, stream>>>(
      simJ, W1, b1, h_a, BK, HID_, 2 * E_, 2 * E_, HID_, HID_, 1);
  layernorm_rows<<<BK, 256, 0, stream>>>(h_a, ln1g, ln1b, HID_);
  gemm_wmma_f32<<<dim3(HID_ / 128, BK / 128), 256, 0, stream>>>(
      h_a, W2, b2, h_b, BK, HID_, HID_, HID_, HID_, HID_, 1);
  layernorm_rows<<<BK, 256, 0, stream>>>(h_b, ln2g, ln2b, HID_);
  rowdot_kernel<<<BK, 256, 0, stream>>>(h_b, Wout, bout, outbk, HID_);

  // 7) concise KL loss
  concise_kernel<<<1, 256, 0, stream>>>(outbk, o);

  // 8) variable-K hard top-k
  khot_kernel<<<B_, 256, 0, stream>>>(outbk, u2, tlen, khot, sel_len);

  // 9) res_mask, mean_selected_sent, ultimate
  zero_f32<<<(B_ * S_) / 256, 256, 0, stream>>>(o_resmask, B_ * S_);
  resmask_scatter<<<(BK + 255) / 256, 256, 0, stream>>>(khot, sel_idx, o_resmask);
  mean_sel_kernel<<<(B_ * E_) / 256, 256, 0, stream>>>(selsent, khot, sel_len, o_meansel);
  ultimate_kernel<<<(BK * E_) / 256, 256, 0, stream>>>(pt, khot, sel_idx, o_ult);

  // 10) anchor / positive projections + unpaired InfoNCE
  small_gemm<<<B_, 256, 0, stream>>>(o_meansel, transW, transb, anc, E_, E_);
  small_gemm<<<B_, 256, 0, stream>>>(meanqa, qaW, qab, posv, E_, E_);
  nce_kernel<<<1, 256, 0, stream>>>(anc, posv, o);

  // 11) redundancy (leave-one-out) + uniformity via Gram matrix
  mss_kernel<<<(KSLOT * E_) / 256, 256, 0, stream>>>(selsent, mss);
  l2norm_rows<<<KSLOT, 256, 0, stream>>>(container, chat, E_);
  l2norm_rows<<<KSLOT, 256, 0, stream>>>(mss, phat, E_);
  gram_kernel<<<KSLOT, 256, 0, stream>>>(chat, G);
  redundancy_kernel<<<1, 256, 0, stream>>>(chat, phat, G, o);
  uniform_kernel<<<1, 256, 0, stream>>>(G, o);
}